// RankingLoss_19335942766645
// MI455X (gfx1250) — compile-verified
//
#include <hip/hip_runtime.h>

typedef __attribute__((ext_vector_type(2))) float v2f;
typedef __attribute__((ext_vector_type(8))) float v8f;

#define CDIM 1000
#define NBLK 256   // 8 wave32s per row

// One block per row: argmax(target_row) -> x1 -> fused relu-margin sum.
// Deterministic: shfl tree within wave32, fixed-order LDS combine across waves.
__global__ __launch_bounds__(NBLK) void rank_row_kernel(
    const float* __restrict__ logits,
    const float* __restrict__ target,
    float* __restrict__ row_loss) {
  const int row = blockIdx.x;
  const int t   = threadIdx.x;
  const size_t roff = (size_t)row * CDIM;
  const float* trow = target + roff;
  const float* lrow = logits + roff;

  // ---- Phase A: argmax of target row (first occurrence wins ties) ----
  float bestv = -__builtin_inff();
  int   besti = CDIM;
  if (t < 250) {  // 250 * float4 == 1000 columns exactly
    float4 tv = *reinterpret_cast<const float4*>(trow + 4 * t);
    float vv[4] = {tv.x, tv.y, tv.z, tv.w};
    bestv = vv[0];
    besti = 4 * t;
#pragma unroll
    for (int k = 1; k < 4; ++k)
      if (vv[k] > bestv) { bestv = vv[k]; besti = 4 * t + k; }
  }
#pragma unroll
  for (int off = 16; off > 0; off >>= 1) {
    float ov = __shfl_down(bestv, off, 32);
    int   oi = __shfl_down(besti, off, 32);
    if (ov > bestv || (ov == bestv && oi < besti)) { bestv = ov; besti = oi; }
  }
  __shared__ float s_v[NBLK / 32];
  __shared__ int   s_i[NBLK / 32];
  __shared__ int   s_label;
  __shared__ float s_x1;
  const int wave = t >> 5, lane = t & 31;
  if (lane == 0) { s_v[wave] = bestv; s_i[wave] = besti; }
  __syncthreads();
  if (t == 0) {
    float bv = s_v[0]; int bi = s_i[0];
#pragma unroll
    for (int w = 1; w < NBLK / 32; ++w)
      if (s_v[w] > bv || (s_v[w] == bv && s_i[w] < bi)) { bv = s_v[w]; bi = s_i[w]; }
    s_label = bi;
    s_x1 = lrow[bi];
  }
  __syncthreads();
  const int   label = s_label;
  const float x1    = s_x1;

  // ---- Phase B: fused relu-margin sum over the logits row ----
  float acc = 0.0f;
  if (label != 0 && t < 250) {  // uniform branch (label is per-row)
    float4 lv4 = *reinterpret_cast<const float4*>(lrow + 4 * t);
    float lv[4] = {lv4.x, lv4.y, lv4.z, lv4.w};
#pragma unroll
    for (int k = 0; k < 4; ++k) {
      const int j = 4 * t + k;
      // j==0 is the negative class: margin == 1.0 exactly (NEG_MARGIN * relu(...))
      const float margin = (j == 0) ? 1.0f : fabsf((float)(label - j)) / 999.0f;
      acc += fmaxf(lv[k] - x1 + margin, 0.0f);
    }
  }
#pragma unroll
  for (int off = 16; off > 0; off >>= 1)
    acc += __shfl_down(acc, off, 32);
  __shared__ float s_s[NBLK / 32];
  if (lane == 0) s_s[wave] = acc;
  __syncthreads();
  if (t == 0) {
    float s = 0.0f;
#pragma unroll
    for (int w = 0; w < NBLK / 32; ++w) s += s_s[w];
    row_loss[row] = s;  // 0.0 when label == 0
  }
}

// Single-wave deterministic mean via V_WMMA_F32_16X16X4_F32:
// D = A x ones  => every column of D holds the 16 row-sums of A.
// Each WMMA reduces one 64-float chunk; 4 independent accumulator chains.
// A 16x4 f32 layout: lane L<16 holds (M=L, K=0..1); lane L>=16 holds (M=L-16, K=2..3)
// => per-lane contiguous float2 at offset 4*(L&15) + 2*(L>>4).
__global__ __launch_bounds__(32) void reduce_mean_wmma(
    const float* __restrict__ vals, float* __restrict__ out, int n) {
  const int lane = threadIdx.x;                 // exactly one wave32, EXEC all 1s
  const int lo = 4 * (lane & 15) + 2 * (lane >> 4);
  const v2f ones = {1.0f, 1.0f};
  v8f c0 = {}; v8f c1 = {}; v8f c2 = {}; v8f c3 = {};
  int base = 0;
  for (; base + 256 <= n; base += 256) {        // uniform trip count
    v2f a0 = *(const v2f*)(vals + base +   0 + lo);
    v2f a1 = *(const v2f*)(vals + base +  64 + lo);
    v2f a2 = *(const v2f*)(vals + base + 128 + lo);
    v2f a3 = *(const v2f*)(vals + base + 192 + lo);
    c0 = __builtin_amdgcn_wmma_f32_16x16x4_f32(false, a0, false, ones, (short)0, c0, false, false);
    c1 = __builtin_amdgcn_wmma_f32_16x16x4_f32(false, a1, false, ones, (short)0, c1, false, false);
    c2 = __builtin_amdgcn_wmma_f32_16x16x4_f32(false, a2, false, ones, (short)0, c2, false, false);
    c3 = __builtin_amdgcn_wmma_f32_16x16x4_f32(false, a3, false, ones, (short)0, c3, false, false);
  }
  v8f c = c0 + c1 + c2 + c3;
  // per-lane: sum over its 8 accumulator VGPRs (M=0..7 or M=8..15 of its column)
  float s = c[0] + c[1] + c[2] + c[3] + c[4] + c[5] + c[6] + c[7];
  // column total = lanes L and L^16 combined; lane 0 holds column 0's full sum
  s += __shfl_xor(s, 16, 32);
  if (lane == 0) {
    for (int i = base; i < n; ++i) s += vals[i];  // tail (empty for n=16384)
    out[0] = s / (float)n;
  }
}

extern "C" void kernel_launch(void* const* d_in, const int* in_sizes, int n_in,
                              void* d_out, int out_size, void* d_ws, size_t ws_size,
                              hipStream_t stream) {
  (void)n_in; (void)out_size; (void)ws_size;
  const float* logits = (const float*)d_in[0];
  const float* target = (const float*)d_in[1];
  float* out = (float*)d_out;
  float* ws  = (float*)d_ws;                    // B floats (64 KB for B=16384)
  const int B = in_sizes[0] / CDIM;
  rank_row_kernel<<<B, NBLK, 0, stream>>>(logits, target, ws);
  reduce_mean_wmma<<<1, 32, 0, stream>>>(ws, out, B);
}